// ChebConv_76003741270359
// MI455X (gfx1250) — compile-verified
//
#include <hip/hip_runtime.h>

#define N_NODES 100000
#define N_EDGES 1600000
#define D 64

typedef __attribute__((ext_vector_type(2))) float v2f;
typedef __attribute__((ext_vector_type(8))) float v8f;

// ---------------------------------------------------------------------------
// Utility: grid-stride zero fill
// ---------------------------------------------------------------------------
__global__ void zero_kernel(float* __restrict__ p, long long n) {
    long long i = (long long)blockIdx.x * blockDim.x + threadIdx.x;
    long long stride = (long long)gridDim.x * blockDim.x;
    for (; i < n; i += stride) p[i] = 0.0f;
}

// ---------------------------------------------------------------------------
// deg[row[e]] += 1
// ---------------------------------------------------------------------------
__global__ void degree_kernel(const int* __restrict__ row, float* __restrict__ deg, int E) {
    int e = blockIdx.x * blockDim.x + threadIdx.x;
    if (e < E) atomicAdd(&deg[row[e]], 1.0f);
}

// dinv = deg > 0 ? deg^-0.5 : 0
__global__ void dinv_kernel(const float* __restrict__ deg, float* __restrict__ dinv, int n) {
    int i = blockIdx.x * blockDim.x + threadIdx.x;
    if (i < n) {
        float d = deg[i];
        dinv[i] = (d > 0.0f) ? (1.0f / sqrtf(d)) : 0.0f;
    }
}

// lap[e] = -dinv[row]*attr*dinv[col]
__global__ void lap_kernel(const int* __restrict__ row, const int* __restrict__ col,
                           const float* __restrict__ attr, const float* __restrict__ dinv,
                           float* __restrict__ lap, int E) {
    int e = blockIdx.x * blockDim.x + threadIdx.x;
    if (e < E) lap[e] = -dinv[row[e]] * attr[e] * dinv[col[e]];
}

// ---------------------------------------------------------------------------
// SPMM scatter: acc[row[e]][j] += lap[e] * h[col[e]][j]
// 64 consecutive threads handle one edge's 64 features (coalesced gather +
// coalesced f32 atomics that resolve in L2 — both h and acc fit in 192MB L2).
// ---------------------------------------------------------------------------
__global__ void spmm_kernel(const int* __restrict__ row, const int* __restrict__ col,
                            const float* __restrict__ lap, const float* __restrict__ h,
                            float* __restrict__ acc, int E) {
    int idx = blockIdx.x * blockDim.x + threadIdx.x;   // E*64 = 102.4M < 2^31
    int e = idx >> 6;
    int j = idx & 63;
    if (e < E) {
        float l = lap[e];
        int c = col[e];
        int r = row[e];
        atomicAdd(&acc[r * D + j], l * h[c * D + j]);
    }
}

// y = 2*y - x   (Chebyshev recurrence epilogue)
__global__ void axpby_kernel(float* __restrict__ y, const float* __restrict__ x, int n) {
    int i = blockIdx.x * blockDim.x + threadIdx.x;
    if (i < n) y[i] = 2.0f * y[i] - x[i];
}

// ---------------------------------------------------------------------------
// WMMA GEMM: out[n,64] (+)= H[n,64] @ W[64,64]  (+ bias on first call)
// One wave per 16x16 C-tile. blockDim = 128 (4 waves); wave w owns column
// tile w*16, the 64x16 B-panel is preloaded into 16 v2f register fragments
// and reused across a grid-stride loop over row tiles.
// ---------------------------------------------------------------------------
__global__ void gemm_wmma_kernel(const float* __restrict__ H, const float* __restrict__ W,
                                 const float* __restrict__ bias, float* __restrict__ out,
                                 int n_row_tiles, int accumulate) {
    const int wave = threadIdx.x >> 5;     // 0..3 -> column tile
    const int lane = threadIdx.x & 31;
    const int col0 = wave * 16;
    const int laneN = lane & 15;           // N (and M for A-frag) within tile
    const int hi = lane >> 4;              // half-wave select (K pairs / M+8)

    // Preload B fragments: B is 4x16 per WMMA step; VGPR0 = K{0|2}, VGPR1 = K{1|3}
    v2f bfrag[16];
#pragma unroll
    for (int ks = 0; ks < 16; ++ks) {
        int k = ks * 4 + hi * 2;
        bfrag[ks].x = W[(k + 0) * D + col0 + laneN];
        bfrag[ks].y = W[(k + 1) * D + col0 + laneN];
    }
    float bval = accumulate ? 0.0f : bias[col0 + laneN];

    for (int tile = blockIdx.x; tile < n_row_tiles; tile += gridDim.x) {
        const int row0 = tile * 16;
        v8f acc;
        if (accumulate) {
#pragma unroll
            for (int r = 0; r < 8; ++r)
                acc[r] = out[(row0 + hi * 8 + r) * D + col0 + laneN];
        } else {
#pragma unroll
            for (int r = 0; r < 8; ++r) acc[r] = bval;
        }

        // A-frag row pointer: lane<16 -> M=laneN, K offset 0; lane>=16 -> K offset 2
        const float* hrow = H + (row0 + laneN) * D + hi * 2;
#pragma unroll
        for (int ks = 0; ks < 16; ++ks) {
            v2f a;
            a.x = hrow[ks * 4 + 0];
            a.y = hrow[ks * 4 + 1];
            acc = __builtin_amdgcn_wmma_f32_16x16x4_f32(
                /*neg_a=*/false, a, /*neg_b=*/false, bfrag[ks],
                /*c_mod=*/(short)0, acc, /*reuse_a=*/false, /*reuse_b=*/false);
        }

#pragma unroll
        for (int r = 0; r < 8; ++r)
            out[(row0 + hi * 8 + r) * D + col0 + laneN] = acc[r];
    }
}

// ---------------------------------------------------------------------------
// Host-side orchestration
// ---------------------------------------------------------------------------
extern "C" void kernel_launch(void* const* d_in, const int* in_sizes, int n_in,
                              void* d_out, int out_size, void* d_ws, size_t ws_size,
                              hipStream_t stream) {
    const float* x         = (const float*)d_in[0];          // [N, 64]
    const int*   edge_idx  = (const int*)d_in[1];            // [2, E] (int32 per harness)
    const float* edge_attr = (const float*)d_in[2];          // [E]
    const float* weight    = (const float*)d_in[3];          // [4, 64, 64]
    const float* bias      = (const float*)d_in[4];          // [64]
    float*       out       = (float*)d_out;                  // [N, 64]

    const int* row = edge_idx;             // edge_index[0]
    const int* col = edge_idx + N_EDGES;   // edge_index[1]

    char* ws = (char*)d_ws;
    float* deg  = (float*)ws; ws += sizeof(float) * N_NODES;
    float* dinv = (float*)ws; ws += sizeof(float) * N_NODES;
    float* lap  = (float*)ws; ws += sizeof(float) * N_EDGES;
    float* A    = (float*)ws; ws += sizeof(float) * (size_t)N_NODES * D;  // tx1
    float* B    = (float*)ws; ws += sizeof(float) * (size_t)N_NODES * D;  // tx2
    float* Z    = (float*)ws; ws += sizeof(float) * (size_t)N_NODES * D;  // tx3

    const int n_row_tiles = N_NODES / 16;        // 6250, exact
    const int GEMM_BLOCKS = 1024;                // 4 waves each (one per col tile)
    const long long spmm_threads = (long long)N_EDGES * D;
    const int spmm_blocks = (int)((spmm_threads + 255) / 256);
    const int e_blocks = (N_EDGES + 255) / 256;
    const int n_blocks = (N_NODES + 255) / 256;
    const int nd_blocks = (N_NODES * D + 255) / 256;

    // 1. zero degree + the three scatter accumulators (A,B,Z are contiguous)
    zero_kernel<<<2048, 256, 0, stream>>>(deg, (long long)N_NODES);
    zero_kernel<<<4096, 256, 0, stream>>>(A, (long long)N_NODES * D * 3);

    // 2. normalization: deg -> dinv -> lap
    degree_kernel<<<e_blocks, 256, 0, stream>>>(row, deg, N_EDGES);
    dinv_kernel<<<n_blocks, 256, 0, stream>>>(deg, dinv, N_NODES);
    lap_kernel<<<e_blocks, 256, 0, stream>>>(row, col, edge_attr, dinv, lap, N_EDGES);

    // 3. out = x @ W0 + bias
    gemm_wmma_kernel<<<GEMM_BLOCKS, 128, 0, stream>>>(x, weight, bias, out, n_row_tiles, 0);

    // 4. tx1 = spmm(x);  out += tx1 @ W1
    spmm_kernel<<<spmm_blocks, 256, 0, stream>>>(row, col, lap, x, A, N_EDGES);
    gemm_wmma_kernel<<<GEMM_BLOCKS, 128, 0, stream>>>(A, weight + 1 * D * D, nullptr, out, n_row_tiles, 1);

    // 5. tx2 = 2*spmm(tx1) - x;  out += tx2 @ W2
    spmm_kernel<<<spmm_blocks, 256, 0, stream>>>(row, col, lap, A, B, N_EDGES);
    axpby_kernel<<<nd_blocks, 256, 0, stream>>>(B, x, N_NODES * D);
    gemm_wmma_kernel<<<GEMM_BLOCKS, 128, 0, stream>>>(B, weight + 2 * D * D, nullptr, out, n_row_tiles, 1);

    // 6. tx3 = 2*spmm(tx2) - tx1;  out += tx3 @ W3
    spmm_kernel<<<spmm_blocks, 256, 0, stream>>>(row, col, lap, B, Z, N_EDGES);
    axpby_kernel<<<nd_blocks, 256, 0, stream>>>(Z, A, N_NODES * D);
    gemm_wmma_kernel<<<GEMM_BLOCKS, 128, 0, stream>>>(Z, weight + 3 * D * D, nullptr, out, n_row_tiles, 1);
}